// LayerController_13082470384336
// MI455X (gfx1250) — compile-verified
//
#include <hip/hip_runtime.h>
#include <hip/hip_bf16.h>

// ---------------- problem constants ----------------
#define N_LAYERS   4
#define N_NODES    128
#define N_FRAMES   512
#define N_SAMPLES  65536
#define N_KEEP     1024
#define DT         (10.0f / 65535.0f)       // jnp.linspace(0,10,65536) step
#define LOG2_10    3.3219280948873623f

// ---------------- workspace layout (bytes) ----------------
#define WS_MEAN 0x0         // 1 float
#define WS_CTRL 0x40        // uint32 ctrl[16]: 0=binHi 1=aboveHi 2=threshKey 3=needEq 4=tieCtr
#define WS_HIST 0x1000      // 65536 u32 (256KB), reused for both radix passes
#define WS_SF   0x50000     // 128*512 f32   (256KB)  sparse frame grid
#define WS_F    0x90000     // 4*128*512 f32 (1MB)    routed forces per layer
#define WS_E    0x190000    // 4*128*512 f32 (1MB)    frame-level energies per layer
#define WS_NC   0x290000    // 4*640 f32: per layer {log2d[128], d128[128], x[128], ten[128], dmp[128]}
#define WS_TMA  0x300000    // 128*65536 f32 (32MB)
#define WS_TMB  0x2300000   // 128*65536 f32 (32MB)

typedef float    v2f __attribute__((ext_vector_type(2)));
typedef float    v8f __attribute__((ext_vector_type(8)));
typedef unsigned v4u __attribute__((ext_vector_type(4)));
typedef int      v4i __attribute__((ext_vector_type(4)));
typedef int      v8i __attribute__((ext_vector_type(8)));

__device__ __forceinline__ v8f wmma_f32(v2f a, v2f b, v8f c) {
  // 8-arg pattern: (neg_a, A, neg_b, B, c_mod, C, reuse_a, reuse_b)
  return __builtin_amdgcn_wmma_f32_16x16x4_f32(false, a, false, b, (short)0, c, false, false);
}

__device__ __forceinline__ float sigm(float x) { return 1.0f / (1.0f + expf(-x)); }

// monotone uint key for float ordering (handles any sign, ties impossible in practice)
__device__ __forceinline__ unsigned fkey(float v) {
  unsigned u = __float_as_uint(v);
  return (u & 0x80000000u) ? ~u : (u | 0x80000000u);
}

// ---------------- stage 0: mean of forces ----------------
__global__ void mean_k(const float* __restrict__ forces, float* __restrict__ meanp) {
  __shared__ float sm[256];
  float s = 0.0f;
  for (int i = threadIdx.x; i < N_SAMPLES; i += 256) s += forces[i];
  sm[threadIdx.x] = s; __syncthreads();
  for (int w = 128; w > 0; w >>= 1) {
    if (threadIdx.x < w) sm[threadIdx.x] += sm[threadIdx.x + w];
    __syncthreads();
  }
  if (threadIdx.x == 0) *meanp = sm[0] * (1.0f / 65536.0f);
}

// ---------------- stage 1: radix select top-1024 ----------------
__global__ void zero_k(unsigned* __restrict__ hist, unsigned* __restrict__ ctrl) {
  int i = blockIdx.x * 256 + threadIdx.x;
  if (i < 65536) hist[i] = 0u;
  if (i < 16) ctrl[i] = 0u;
}

__global__ void hist_hi_k(const float* __restrict__ forces, const float* __restrict__ meanp,
                          unsigned* __restrict__ hist) {
  int i = blockIdx.x * 256 + threadIdx.x;
  unsigned key = fkey(forces[i] + *meanp);
  atomicAdd(&hist[key >> 16], 1u);
}

__global__ void select_hi_k(unsigned* __restrict__ hist, unsigned* __restrict__ ctrl) {
  if (threadIdx.x == 0) {
    unsigned cum = 0;
    for (int b = 65535; b >= 0; --b) {
      unsigned h = hist[b];
      if (cum + h >= N_KEEP) { ctrl[0] = (unsigned)b; ctrl[1] = cum; break; }
      cum += h;
    }
  }
  __syncthreads();
  for (int i = threadIdx.x; i < 65536; i += 256) hist[i] = 0u;   // reuse for lo pass
}

__global__ void hist_lo_k(const float* __restrict__ forces, const float* __restrict__ meanp,
                          unsigned* __restrict__ hist, const unsigned* __restrict__ ctrl) {
  int i = blockIdx.x * 256 + threadIdx.x;
  unsigned key = fkey(forces[i] + *meanp);
  if ((key >> 16) == ctrl[0]) atomicAdd(&hist[key & 0xFFFFu], 1u);
}

__global__ void select_lo_k(const unsigned* __restrict__ hist, unsigned* __restrict__ ctrl) {
  if (threadIdx.x == 0) {
    unsigned need = N_KEEP - ctrl[1];
    unsigned cum = 0;
    for (int b = 65535; b >= 0; --b) {
      unsigned h = hist[b];
      if (cum + h >= need) {
        ctrl[2] = (ctrl[0] << 16) | (unsigned)b;  // exact threshold key
        ctrl[3] = need - cum;                     // how many ==threshold to keep
        ctrl[4] = 0u;                             // tie counter
        break;
      }
      cum += h;
    }
  }
}

__global__ void build_sf_k(const float* __restrict__ forces, const float* __restrict__ meanp,
                           unsigned* __restrict__ ctrl, float* __restrict__ sf) {
  int i = blockIdx.x * 256 + threadIdx.x;   // i = n*512 + j (matches forces layout)
  float v = forces[i] + *meanp;
  unsigned key = fkey(v);
  bool keep = key > ctrl[2];
  if (!keep && key == ctrl[2]) keep = (atomicAdd(&ctrl[4], 1u) < ctrl[3]);
  sf[i] = keep ? v : 0.0f;
}

// ---------------- stage 2: force routing GEMM (WMMA fp32), all 4 layers ----------------
// F[l][d][j] = sum_n R_l[n][d] * SF[n][j],  M=128(d) K=128(n) N=512(j)
__global__ void force_gemm_k(const float* __restrict__ fr, const float* __restrict__ sf,
                             float* __restrict__ F) {
  int wave = blockIdx.x * 4 + (threadIdx.x >> 5);
  int lane = threadIdx.x & 31;
  int l  = wave >> 8;
  int mt = (wave >> 5) & 7;
  int nt = wave & 31;
  int d0 = mt * 16, j0 = nt * 16;
  const float* R = fr + l * N_NODES * N_NODES;
  int mr = lane & 15, hk = lane >> 4, kk = hk * 2;
  v8f c = {};
  for (int k0 = 0; k0 < 128; k0 += 4) {
    v2f a, b;
    a.x = R[(k0 + kk)     * 128 + d0 + mr];
    a.y = R[(k0 + kk + 1) * 128 + d0 + mr];
    b.x = sf[(k0 + kk)     * 512 + j0 + mr];
    b.y = sf[(k0 + kk + 1) * 512 + j0 + mr];
    c = wmma_f32(a, b, c);
  }
  int n = lane & 15;
  float* Fl = F + (size_t)l * 128 * 512;
#pragma unroll
  for (int r = 0; r < 8; ++r) {
    int m = r + 8 * hk;
    Fl[(d0 + m) * 512 + j0 + n] = c[r];
  }
}

// ---------------- stage 3: per-node constants ----------------
__global__ void node_const_k(const float* __restrict__ damp, const float* __restrict__ mass,
                             const float* __restrict__ tension, float* __restrict__ nc) {
  int i = blockIdx.x * 256 + threadIdx.x;
  if (i >= N_LAYERS * N_NODES) return;
  int l = i >> 7, d = i & 127;
  float dmp = 0.9997f + sigm(damp[l * 128 + d]) * 0.0002f;
  float l2  = log2f(dmp);
  float ms  = sigm(mass[l * 128 + d]) * 500.0f;
  float* b  = nc + l * 640;
  b[0 * 128 + d] = l2;                  // log2(damp)
  b[1 * 128 + d] = exp2f(128.0f * l2);  // damp^128
  b[2 * 128 + d] = 1.0f / (2.0f * ms);  // x = damping/(2*mass), damping==1
  b[3 * 128 + d] = tension[l * 128 + d];
  b[4 * 128 + d] = dmp;
}

// ---------------- stage 4: 512-step frame scan per (layer,node) ----------------
// E_j = d^128 * E_{j-1} + d * F_j
__global__ void scan_k(const float* __restrict__ F, float* __restrict__ E,
                       const float* __restrict__ nc) {
  int i = blockIdx.x * 256 + threadIdx.x;
  if (i >= N_LAYERS * N_NODES) return;
  int l = i >> 7, d = i & 127;
  const float* b = nc + l * 640;
  float dmp = b[4 * 128 + d], d128 = b[1 * 128 + d];
  const float* f = F + (size_t)(l * 128 + d) * 512;
  float*       e = E + (size_t)(l * 128 + d) * 512;
  float acc = 0.0f;
  for (int j = 0; j < 512; ++j) { acc = d128 * acc + dmp * f[j]; e[j] = acc; }
}

// ---------------- stage 5: layer 0 oscillator (tension constant per node) ----------------
// a*cos(wt-phi) == cos(wt) + (x/w)*sin(wt)  (exact identity, kills atan2/1/cos)
__global__ void layer0_osc_k(float* __restrict__ tm, const float* __restrict__ nc,
                             const float* __restrict__ E) {
  size_t idx = (size_t)blockIdx.x * 256 + threadIdx.x;   // d*65536 + s
  int d = (int)(idx >> 16), s = (int)(idx & 65535);
  float x     = nc[2 * 128 + d];
  float T10   = exp2f(nc[3 * 128 + d] * LOG2_10);
  float omega = sqrtf(fabsf(T10 - x * x));
  float energy = E[d * 512 + (s >> 7)] * exp2f((float)(s & 127) * nc[d]);
  float sn, cs; __sincosf(omega * (s * DT), &sn, &cs);
  tm[idx] = energy * (cs + (x / omega) * sn);
}

// ---------------- stage 6: fused tension-GEMM + oscillator (layers 1..3) ----------------
// One block (8 waves) owns a 16-sample column strip: the 128x16 B panel of tm is DMA'd
// into LDS once via the Tensor Data Mover, then all 8 waves (one per 16-row M tile)
// consume it from LDS. tq[d][s] = sum_n TR[n][d]*tmin[n][s]; oscillator fused in epilogue.
__global__ void __launch_bounds__(256)
fused_gemm_osc_k(const float* __restrict__ TR, const float* __restrict__ tmin,
                 float* __restrict__ tmout, const float* __restrict__ nc,
                 const float* __restrict__ E) {
  __shared__ float Bs[128 * 16];          // B panel [k][n], 8KB
  const int s0   = blockIdx.x * 16;
  const int lane = threadIdx.x & 31;
  const int waveId = threadIdx.x >> 5;

  // --- TDM: stage B panel (tile 16 wide x 128 rows, row stride 65536 elems) ---
  if (waveId == 0) {
    unsigned ldsAddr = (unsigned)(size_t)(void*)Bs;   // low 32 bits of flat LDS addr = LDS offset
    unsigned long long ga = (unsigned long long)(size_t)(tmin + s0);
    v4u g0;
    g0.x = 1u;                                        // count=1, is_restore=0, gather off
    g0.y = ldsAddr;                                   // lds_addr [63:32]
    g0.z = (unsigned)ga;                              // global_addr [95:64]
    g0.w = ((unsigned)(ga >> 32) & 0x01FFFFFFu) | (2u << 30);  // addr[56:32] | type=2
    v8i g1;
    g1[0] = 0x00020000;                               // wg_mask=0, data_size=2 (4B)
    g1[1] = (int)((65536u & 0xFFFFu) << 16);          // tensor_dim0[15:0] << 16  (== 0)
    g1[2] = (int)((65536u >> 16) | (128u << 16));     // tensor_dim0[31:16] | tensor_dim1[15:0]
    g1[3] = (int)(16u << 16);                         // tensor_dim1[31:16] | tile_dim0=16
    g1[4] = 128;                                      // tile_dim1=128, tile_dim2=0
    g1[5] = 65536;                                    // tensor_dim0_stride[31:0]
    g1[6] = 0;                                        // stride hi16 | dim1_stride lo16
    g1[7] = 0;
    v4i gz = {0, 0, 0, 0};
#if __clang_major__ >= 23
    v8i gz8 = {0, 0, 0, 0, 0, 0, 0, 0};
    __builtin_amdgcn_tensor_load_to_lds(g0, g1, gz, gz, gz8, 0);
#else
    __builtin_amdgcn_tensor_load_to_lds(g0, g1, gz, gz, 0);
#endif
    __builtin_amdgcn_s_wait_tensorcnt(0);
  }
  __syncthreads();

  // --- per-wave 16x16 tile: M tile = waveId, all share the LDS B panel ---
  const int d0 = waveId * 16;
  const int mr = lane & 15, hk = lane >> 4, kk = hk * 2;
  v8f c = {};
  for (int k0 = 0; k0 < 128; k0 += 4) {
    v2f a, b;
    a.x = TR[(k0 + kk)     * 128 + d0 + mr];
    a.y = TR[(k0 + kk + 1) * 128 + d0 + mr];
    b.x = Bs[(k0 + kk)     * 16 + mr];
    b.y = Bs[(k0 + kk + 1) * 16 + mr];
    c = wmma_f32(a, b, c);
  }

  const int n = lane & 15, s = s0 + n;
  const int j = s >> 7;
  const float r = (float)(s & 127);
  const float t = s * DT;
#pragma unroll
  for (int rr = 0; rr < 8; ++rr) {
    int d = d0 + rr + 8 * hk;
    float x     = nc[2 * 128 + d];
    float T10   = exp2f((nc[3 * 128 + d] + c[rr]) * LOG2_10);
    float omega = sqrtf(fabsf(T10 - x * x));
    float energy = E[d * 512 + j] * exp2f(r * nc[d]);
    float sn, cs; __sincosf(omega * t, &sn, &cs);
    tmout[(size_t)d * N_SAMPLES + s] = energy * (cs + (x / omega) * sn);
  }
}

// ---------------- stage 7: reduce over nodes ----------------
__global__ void reduce_k(const float* __restrict__ tm, float* __restrict__ out) {
  int s = blockIdx.x * 256 + threadIdx.x;
  float acc = 0.0f;
#pragma unroll 4
  for (int d = 0; d < N_NODES; ++d) acc += tm[(size_t)d * N_SAMPLES + s];
  out[s] = acc;
}

// ---------------- host launcher ----------------
extern "C" void kernel_launch(void* const* d_in, const int* in_sizes, int n_in,
                              void* d_out, int out_size, void* d_ws, size_t ws_size,
                              hipStream_t stream) {
  (void)in_sizes; (void)n_in; (void)out_size; (void)ws_size;
  const float* forces  = (const float*)d_in[0];   // [1,128,512]
  const float* damp    = (const float*)d_in[1];   // [4,1,128,1]
  const float* mass    = (const float*)d_in[2];   // [4,1,128,1]
  const float* tension = (const float*)d_in[3];   // [4,1,128,1]
  const float* frout   = (const float*)d_in[4];   // [4,128,128]
  const float* trout   = (const float*)d_in[5];   // [4,128,128]
  float* out = (float*)d_out;                     // [1,1,65536]

  char* ws = (char*)d_ws;
  float*    meanp = (float*)(ws + WS_MEAN);
  unsigned* ctrl  = (unsigned*)(ws + WS_CTRL);
  unsigned* hist  = (unsigned*)(ws + WS_HIST);
  float*    sf    = (float*)(ws + WS_SF);
  float*    F     = (float*)(ws + WS_F);
  float*    E     = (float*)(ws + WS_E);
  float*    nc    = (float*)(ws + WS_NC);
  float*    tmA   = (float*)(ws + WS_TMA);
  float*    tmB   = (float*)(ws + WS_TMB);

  // sparsification pipeline
  mean_k<<<1, 256, 0, stream>>>(forces, meanp);
  zero_k<<<256, 256, 0, stream>>>(hist, ctrl);
  hist_hi_k<<<256, 256, 0, stream>>>(forces, meanp, hist);
  select_hi_k<<<1, 256, 0, stream>>>(hist, ctrl);
  hist_lo_k<<<256, 256, 0, stream>>>(forces, meanp, hist, ctrl);
  select_lo_k<<<1, 1, 0, stream>>>(hist, ctrl);
  build_sf_k<<<256, 256, 0, stream>>>(forces, meanp, ctrl, sf);

  // force routing for all layers: 1024 tiles, 4 waves/block
  force_gemm_k<<<256, 128, 0, stream>>>(frout, sf, F);

  // per-node constants + frame scans (all layers, independent)
  node_const_k<<<2, 256, 0, stream>>>(damp, mass, tension, nc);
  scan_k<<<2, 256, 0, stream>>>(F, E, nc);

  // layer 0: constant-tension oscillator
  layer0_osc_k<<<(N_NODES * N_SAMPLES) / 256, 256, 0, stream>>>(tmA, nc, E);

  // layers 1..3: fused WMMA GEMM + oscillator, one block per 16-sample strip
  const int gemmBlocks = N_SAMPLES / 16;   // 4096 blocks x 8 waves
  fused_gemm_osc_k<<<gemmBlocks, 256, 0, stream>>>(trout + 1 * 16384, tmA, tmB,
                                                   nc + 1 * 640, E + 1 * 128 * 512);
  fused_gemm_osc_k<<<gemmBlocks, 256, 0, stream>>>(trout + 2 * 16384, tmB, tmA,
                                                   nc + 2 * 640, E + 2 * 128 * 512);
  fused_gemm_osc_k<<<gemmBlocks, 256, 0, stream>>>(trout + 3 * 16384, tmA, tmB,
                                                   nc + 3 * 640, E + 3 * 128 * 512);

  // final node-sum
  reduce_k<<<N_SAMPLES / 256, 256, 0, stream>>>(tmB, out);
}